// downprompt_9569187136133
// MI455X (gfx1250) — compile-verified
//
#include <hip/hip_runtime.h>
#include <hip/hip_bf16.h>
#include <math.h>

typedef __attribute__((ext_vector_type(2))) float v2f;
typedef __attribute__((ext_vector_type(8))) float v8f;

#define FT 256
#define NC 16
#define EPSV 1e-8f

// workspace layout (in floats)
#define WS_C      0        // c[256]
#define WS_SUMS   256      // sums[16][256]   (atomic-fallback path)
#define WS_COUNTS 4352     // counts[16]
#define WS_AVET   4368     // aveT[256][16]
#define WS_ANORM  8464     // aveNorm[16]
#define WS_PART   8480     // per-block partials: nPart * 4112
#define PART_STRIDE (NC*FT + NC)   // 4112 floats per block
#define SEG_BLOCKS 256

__device__ __forceinline__ float eluf(float x) {
    return x > 0.f ? x : (__expf(x) - 1.f);
}

// ---------------------------------------------------------------- prep
// c[f] = w_wf0 * (1 + elu(sum_j w_wp[j]*prompt_j[f])) + w_wf1 * w_dp[f]
__global__ void dp_prep_kernel(const float* __restrict__ p1, const float* __restrict__ p2,
                               const float* __restrict__ p3, const float* __restrict__ wwp,
                               const float* __restrict__ wwf, const float* __restrict__ wdp,
                               float* __restrict__ ws) {
    int f = threadIdx.x;  // 256 threads
    float s = wwp[0]*p1[f] + wwp[1]*p2[f] + wwp[2]*p3[f];
    float weight = 1.f + eluf(s);
    ws[WS_C + f] = wwf[0]*weight + wwf[1]*wdp[f];
    for (int cl = 0; cl < NC; ++cl) ws[WS_SUMS + cl*FT + f] = 0.f;
    if (f < NC) ws[WS_COUNTS + f] = 0.f;
}

// ---------------------------------------------------------------- segment sums
// Each thread owns feature column t -> LDS accumulation without atomics.
// Deterministic variant: each block writes its partial to a private ws slot.
__global__ void dp_segsum_part_kernel(const float* __restrict__ seq, const float* __restrict__ seq1,
                                      const int* __restrict__ labels, const float* __restrict__ a4p,
                                      float* __restrict__ ws, int n, int rowsPerBlock) {
    __shared__ float ps[NC*FT];
    __shared__ float pc[NC];
    int t = threadIdx.x;  // 256
    for (int i = t; i < NC*FT; i += FT) ps[i] = 0.f;
    if (t < NC) pc[t] = 0.f;
    __syncthreads();
    float cc = ws[WS_C + t];
    float a4 = a4p[0];
    int r0 = blockIdx.x * rowsPerBlock;
    int r1 = r0 + rowsPerBlock; if (r1 > n) r1 = n;
    for (int r = r0; r < r1; ++r) {
        int lb = labels[r];
        float s  = seq [(size_t)r*FT + t];
        float s1 = seq1[(size_t)r*FT + t];
        float v = eluf(cc*s) + a4*s1;
        ps[lb*FT + t] += v;
        if (t == 0) pc[lb] += 1.f;
    }
    __syncthreads();
    float* dst = ws + WS_PART + (size_t)blockIdx.x * PART_STRIDE;
    for (int i = t; i < NC*FT; i += FT) dst[i] = ps[i];
    if (t < NC) dst[NC*FT + t] = pc[t];
}

// Atomic fallback (used if ws too small for deterministic partials, or if 0 < n < SEG_BLOCKS)
__global__ void dp_segsum_atomic_kernel(const float* __restrict__ seq, const float* __restrict__ seq1,
                                        const int* __restrict__ labels, const float* __restrict__ a4p,
                                        float* __restrict__ ws, int n, int rowsPerBlock) {
    __shared__ float ps[NC*FT];
    __shared__ float pc[NC];
    int t = threadIdx.x;
    for (int i = t; i < NC*FT; i += FT) ps[i] = 0.f;
    if (t < NC) pc[t] = 0.f;
    __syncthreads();
    float cc = ws[WS_C + t];
    float a4 = a4p[0];
    int r0 = blockIdx.x * rowsPerBlock;
    int r1 = r0 + rowsPerBlock; if (r1 > n) r1 = n;
    for (int r = r0; r < r1; ++r) {
        int lb = labels[r];
        float s  = seq [(size_t)r*FT + t];
        float s1 = seq1[(size_t)r*FT + t];
        float v = eluf(cc*s) + a4*s1;
        ps[lb*FT + t] += v;
        if (t == 0) pc[lb] += 1.f;
    }
    __syncthreads();
    for (int i = t; i < NC*FT; i += FT) atomicAdd(&ws[WS_SUMS + i], ps[i]);
    if (t < NC) atomicAdd(&ws[WS_COUNTS + t], pc[t]);
}

// ---------------------------------------------------------------- class means + norms
// nPart>0: deterministic reduction of per-block partials; nPart==0: read atomics result.
__global__ void dp_ave_kernel(float* __restrict__ ws, int nPart) {
    __shared__ float sums_sh[NC*FT];
    __shared__ float cnt_sh[NC];
    int t = threadIdx.x;  // 256
    if (nPart > 0) {
        for (int cl = 0; cl < NC; ++cl) {
            float acc = 0.f;
            for (int b = 0; b < nPart; ++b)
                acc += ws[WS_PART + (size_t)b*PART_STRIDE + cl*FT + t];
            sums_sh[cl*FT + t] = acc;
        }
        if (t < NC) {
            float acc = 0.f;
            for (int b = 0; b < nPart; ++b)
                acc += ws[WS_PART + (size_t)b*PART_STRIDE + NC*FT + t];
            cnt_sh[t] = acc;
        }
    } else {
        for (int i = t; i < NC*FT; i += FT) sums_sh[i] = ws[WS_SUMS + i];
        if (t < NC) cnt_sh[t] = ws[WS_COUNTS + t];
    }
    __syncthreads();
    // aveT[f][n] = sums[n][f] / max(count[n],1)  (zero-count classes stay 0)
    int f = t;
    for (int n2 = 0; n2 < NC; ++n2) {
        float inv = 1.f / fmaxf(cnt_sh[n2], 1.f);
        ws[WS_AVET + f*NC + n2] = sums_sh[n2*FT + f] * inv;
    }
    if (t < NC) {
        float inv = 1.f / fmaxf(cnt_sh[t], 1.f);
        float ss = 0.f;
        for (int f2 = 0; f2 < FT; ++f2) {
            float v = sums_sh[t*FT + f2] * inv;
            ss += v*v;
        }
        ws[WS_ANORM + t] = sqrtf(ss);
    }
}

// ---------------------------------------------------------------- WMMA scores + softmax
#define TPB 2          // 16-row tiles per block (1 per wave32)
#define TSTRIDE 260    // padded row stride (floats) -> bank-conflict free

__global__ void dp_wmma_kernel(const float* __restrict__ seq, const float* __restrict__ seq1,
                               const float* __restrict__ a4p, const float* __restrict__ ws,
                               float* __restrict__ out, int n, int numTiles) {
    __shared__ float c_sh[FT];
    __shared__ float aveT_sh[FT*NC];        // B-matrix: aveT[k][n]
    __shared__ float an_sh[NC];
    __shared__ float tiles_sh[TPB*16*TSTRIDE];
    __shared__ float scores_sh[TPB*256];

    int t = threadIdx.x;  // 64 threads = 2 waves
    for (int i = t; i < FT;    i += 64) c_sh[i]    = ws[WS_C + i];
    for (int i = t; i < FT*NC; i += 64) aveT_sh[i] = ws[WS_AVET + i];
    if (t < NC) an_sh[t] = ws[WS_ANORM + t];
    __syncthreads();

    int w = t >> 5, lane = t & 31;
    int tile = blockIdx.x * TPB + w;
    bool active = (tile < numTiles);        // wave-uniform
    float a4 = a4p[0];
    float* myTile = tiles_sh + w*16*TSTRIDE;
    int m  = lane & 15;                     // row in tile / column index
    int kh = lane >> 4;                     // K-half selector (A/B frag high lanes)

    // ---- stage rawret tile (coalesced float4 global loads) ----
    if (active) {
        int rowBase = tile * 16;
        for (int r = 0; r < 16; ++r) {
            int row = rowBase + r;
            if (row < n) {
                const float4* psq  = (const float4*)(seq  + (size_t)row*FT);
                const float4* psq1 = (const float4*)(seq1 + (size_t)row*FT);
                #pragma unroll
                for (int chunk = 0; chunk < 2; ++chunk) {
                    int i4 = chunk*32 + lane;
                    float4 s  = psq[i4];
                    float4 s1 = psq1[i4];
                    int f = i4*4;
                    float4 rv;
                    rv.x = eluf(c_sh[f+0]*s.x) + a4*s1.x;
                    rv.y = eluf(c_sh[f+1]*s.y) + a4*s1.y;
                    rv.z = eluf(c_sh[f+2]*s.z) + a4*s1.z;
                    rv.w = eluf(c_sh[f+3]*s.w) + a4*s1.w;
                    *(float4*)(myTile + r*TSTRIDE + f) = rv;
                }
            } else {
                #pragma unroll
                for (int chunk = 0; chunk < 2; ++chunk) {
                    int f = (chunk*32 + lane)*4;
                    float4 z; z.x = z.y = z.z = z.w = 0.f;
                    *(float4*)(myTile + r*TSTRIDE + f) = z;
                }
            }
        }
    }
    __syncthreads();

    // ---- row L2 norms: lane handles row m, K-half kh; combine via shfl_xor(16) ----
    float rowNorm = 0.f;
    if (active) {
        const float4* rp = (const float4*)(myTile + m*TSTRIDE + kh*128);
        float ss = 0.f;
        #pragma unroll 8
        for (int j = 0; j < 32; ++j) {
            float4 v = rp[j];
            ss += v.x*v.x + v.y*v.y + v.z*v.z + v.w*v.w;
        }
        ss += __shfl_xor(ss, 16, 32);
        rowNorm = sqrtf(ss);                // valid in lanes 0..15 for row=lane
    }

    // ---- K-loop: 64 x V_WMMA_F32_16X16X4_F32, D = A(16x4) * B(4x16) + C ----
    v8f acc = {0.f,0.f,0.f,0.f,0.f,0.f,0.f,0.f};
    if (active) {
        #pragma unroll 8
        for (int k0 = 0; k0 < FT; k0 += 4) {
            int kb = k0 + 2*kh;             // low lanes K0/K1, high lanes K2/K3
            const float* ap = myTile + m*TSTRIDE + kb;
            v2f a; a.x = ap[0];             a.y = ap[1];
            v2f b; b.x = aveT_sh[kb*NC + m]; b.y = aveT_sh[(kb+1)*NC + m];
            acc = __builtin_amdgcn_wmma_f32_16x16x4_f32(
                false, a, false, b, (short)0, acc, false, false);
        }
        // C/D layout: VGPR i -> M = i + 8*kh, N = lane&15
        float* sc = scores_sh + w*256;
        #pragma unroll
        for (int i = 0; i < 8; ++i) sc[(i + 8*kh)*NC + m] = acc[i];
    }
    __syncthreads();

    // ---- cosine division + softmax (lanes 0..15, one row each) ----
    if (active && lane < 16) {
        int row = tile*16 + lane;
        if (row < n) {
            const float* sc = scores_sh + w*256 + lane*NC;
            float r[NC];
            float mx = -1e30f;
            #pragma unroll
            for (int j = 0; j < NC; ++j) {
                float denom = fmaxf(rowNorm * an_sh[j], EPSV);
                r[j] = sc[j] / denom;
                mx = fmaxf(mx, r[j]);
            }
            float sum = 0.f;
            #pragma unroll
            for (int j = 0; j < NC; ++j) { r[j] = __expf(r[j] - mx); sum += r[j]; }
            float inv = 1.f / sum;
            float4* op = (float4*)(out + (size_t)row*NC);
            #pragma unroll
            for (int j = 0; j < 4; ++j) {
                float4 o;
                o.x = r[4*j+0]*inv; o.y = r[4*j+1]*inv;
                o.z = r[4*j+2]*inv; o.w = r[4*j+3]*inv;
                op[j] = o;
            }
        }
    }
}

// ---------------------------------------------------------------- launch
extern "C" void kernel_launch(void* const* d_in, const int* in_sizes, int n_in,
                              void* d_out, int out_size, void* d_ws, size_t ws_size,
                              hipStream_t stream) {
    const float* seq    = (const float*)d_in[0];
    const float* seq1   = (const float*)d_in[1];
    const int*   labels = (const int*)  d_in[2];
    const float* p1     = (const float*)d_in[3];
    const float* p2     = (const float*)d_in[4];
    const float* p3     = (const float*)d_in[5];
    const float* wwp    = (const float*)d_in[6];
    const float* wwf    = (const float*)d_in[7];
    const float* wdp    = (const float*)d_in[8];
    const float* a4     = (const float*)d_in[9];
    float* ws  = (float*)d_ws;
    float* out = (float*)d_out;

    int n = in_sizes[0] / FT;   // number of rows (N)

    dp_prep_kernel<<<1, FT, 0, stream>>>(p1, p2, p3, wwp, wwf, wdp, ws);

    size_t needDet = (size_t)(WS_PART + (size_t)SEG_BLOCKS * PART_STRIDE) * sizeof(float);
    int grid = SEG_BLOCKS;
    if (grid > n && n > 0) grid = n;
    if (grid < 1) grid = 1;
    int rpb = (n + grid - 1) / grid;

    if (ws_size >= needDet && grid == SEG_BLOCKS) {
        dp_segsum_part_kernel<<<grid, FT, 0, stream>>>(seq, seq1, labels, a4, ws, n, rpb);
        dp_ave_kernel<<<1, FT, 0, stream>>>(ws, grid);
    } else {
        dp_segsum_atomic_kernel<<<grid, FT, 0, stream>>>(seq, seq1, labels, a4, ws, n, rpb);
        dp_ave_kernel<<<1, FT, 0, stream>>>(ws, 0);
    }

    int numTiles = (n + 15) / 16;
    int gridD = (numTiles + TPB - 1) / TPB;
    dp_wmma_kernel<<<gridD, 64, 0, stream>>>(seq, seq1, a4, ws, out, n, numTiles);
}